// NAM_20126216750029
// MI455X (gfx1250) — compile-verified
//
#include <hip/hip_runtime.h>
#include <hip/hip_bf16.h>

#define FDIM   512
#define HDIM   32
#define BATCH  8192
#define ROWS   32          // batch rows per block
#define THREADS 256        // 8 wave32s

// LDS layout (float indices). Pad each feature's 32 weights to 36 floats:
// stride 144 B keeps B128 destinations 16B-aligned and limits ds_load bank
// conflicts to 2-way on the one-time weight reads.
#define PART_OFF   0                         // 8 waves x 32 rows partial sums
#define ARR_STRIDE 36
#define ARR_FLOATS (FDIM * ARR_STRIDE)       // 18432
#define W1_OFF     256
#define B1_OFF     (W1_OFF + ARR_FLOATS)
#define W2_OFF     (B1_OFF + ARR_FLOATS)
#define SMEM_FLOATS (W2_OFF + ARR_FLOATS)    // 55552 floats
#define SMEM_BYTES  (SMEM_FLOATS * 4)        // 222208 B  (< 320 KB WGP LDS)

typedef int v4i __attribute__((ext_vector_type(4)));
typedef __attribute__((address_space(1))) v4i* global_v4i_ptr;
typedef __attribute__((address_space(3))) v4i* lds_v4i_ptr;
typedef float v2f __attribute__((ext_vector_type(2)));
typedef float v8f __attribute__((ext_vector_type(8)));

__device__ __forceinline__ void async_load_b128(const float* gsrc, float* lds_dst) {
#if __has_builtin(__builtin_amdgcn_global_load_async_to_lds_b128)
    __builtin_amdgcn_global_load_async_to_lds_b128(
        (global_v4i_ptr)gsrc,
        (lds_v4i_ptr)lds_dst,
        0, 0);
#else
    unsigned int loff = (unsigned int)(unsigned long long)lds_dst; // LDS byte offset (addr[31:0])
    asm volatile("global_load_async_to_lds_b128 %0, %1, off"
                 :: "v"(loff), "v"(gsrc) : "memory");
#endif
}

__device__ __forceinline__ void wait_async0() {
#if __has_builtin(__builtin_amdgcn_s_wait_asynccnt)
    __builtin_amdgcn_s_wait_asynccnt(0);
#else
    asm volatile("s_wait_asynccnt 0" ::: "memory");
#endif
}

extern "C" __global__ void __launch_bounds__(THREADS)
NAM_20126216750029_kernel(const float* __restrict__ x,
                          const float* __restrict__ w1,
                          const float* __restrict__ b1,
                          const float* __restrict__ w2,
                          const float* __restrict__ b2,
                          float* __restrict__ out,
                          float* __restrict__ contribs)
{
    extern __shared__ float smem[];
    const int tid  = threadIdx.x;
    const int wave = tid >> 5;
    const int lane = tid & 31;
    const int row0 = blockIdx.x * ROWS;

    // Zero the 8x32 per-(wave,row) partial-sum slots.
    smem[PART_OFF + tid] = 0.0f;

    // ---- Async stage w1,b1,w2 (each FDIM*HDIM f32) into padded LDS ----
    const float* srcs[3] = { w1, b1, w2 };
    const int    offs[3] = { W1_OFF, B1_OFF, W2_OFF };
    #pragma unroll
    for (int a = 0; a < 3; ++a) {
        const float* g  = srcs[a];
        const int    ao = offs[a];
        #pragma unroll
        for (int i = 0; i < (FDIM * HDIM / 4) / THREADS; ++i) {  // 16 iters
            int v  = tid + i * THREADS;       // vec4 index 0..4095
            int f  = v >> 3;                  // 8 vec4s per feature
            int h0 = (v & 7) << 2;
            async_load_b128(g + (size_t)v * 4, &smem[ao + f * ARR_STRIDE + h0]);
        }
    }
    wait_async0();
    __syncthreads();

    // ---- Compute: wave w owns features [64w, 64w+64) as two 32-lane chunks ----
    #pragma unroll
    for (int ci = 0; ci < 2; ++ci) {
        const int f    = wave * 64 + ci * 32 + lane;   // this lane's feature
        const int base = f * ARR_STRIDE;

        float rw1[HDIM], rb1[HDIM], rw2[HDIM];
        #pragma unroll
        for (int h = 0; h < HDIM; ++h) {
            rw1[h] = smem[W1_OFF + base + h];
            rb1[h] = smem[B1_OFF + base + h];
            rw2[h] = smem[W2_OFF + base + h];
        }
        const float bb2 = b2[f];

        for (int r = 0; r < ROWS; ++r) {
            const int row = row0 + r;
            const float xv = x[(size_t)row * FDIM + f];   // coalesced 128B/wave

            float acc = bb2;
            #pragma unroll
            for (int h = 0; h < HDIM; ++h) {
                float t = __builtin_fmaf(xv, rw1[h], rb1[h]);
                t = fmaxf(t, 0.0f);                        // ReLU
                acc = __builtin_fmaf(t, rw2[h], acc);
            }
            contribs[(size_t)row * FDIM + f] = acc;        // coalesced store

            // wave32 reduction of this chunk's 32 feature contribs
            float s = acc;
            #pragma unroll
            for (int o = 16; o > 0; o >>= 1) s += __shfl_xor(s, o, 32);
            if (lane == 0) {
                float* p = &smem[PART_OFF + wave * ROWS + r];
                *p = (ci == 0) ? s : (*p + s);   // same lane both chunks: no race
            }
        }
    }
    __syncthreads();

    // ---- out[row] = column sums of the 8x32 partial matrix, via WMMA ----
    // D = A(ones,16x4) x B(4x16 partials) + C  =>  every row of D holds the
    // 4-wave column sums; chain two WMMAs to cover all 8 waves. Column sums
    // are invariant to the K <-> (VGPR, lane-half) mapping, so this is
    // layout-robust; only N = lane&15 matters (documented for C/D VGPR0).
    if (wave < 2) {                       // wave w handles rows [16w, 16w+16)
        const int half = wave;
        const int n    = lane & 15;
        const int kk   = (lane >> 4) << 1;    // 0 or 2: this lane-half's K rows
        v2f ones; ones.x = 1.0f; ones.y = 1.0f;
        v8f acc = {};
        #pragma unroll
        for (int g = 0; g < 2; ++g) {         // wave groups 0-3, 4-7
            v2f bmat;
            bmat.x = smem[PART_OFF + (g * 4 + kk    ) * ROWS + half * 16 + n];
            bmat.y = smem[PART_OFF + (g * 4 + kk + 1) * ROWS + half * 16 + n];
            acc = __builtin_amdgcn_wmma_f32_16x16x4_f32(
                /*neg_a=*/false, ones, /*neg_b=*/false, bmat,
                /*c_mod=*/(short)0, acc, /*reuse_a=*/false, /*reuse_b=*/false);
        }
        if (lane < 16)                        // D VGPR0, lanes 0-15: (M=0, N=lane)
            out[row0 + half * 16 + lane] = acc[0];
    }
}

extern "C" void kernel_launch(void* const* d_in, const int* in_sizes, int n_in,
                              void* d_out, int out_size, void* d_ws, size_t ws_size,
                              hipStream_t stream) {
    (void)in_sizes; (void)n_in; (void)out_size; (void)d_ws; (void)ws_size;
    const float* x  = (const float*)d_in[0];   // [B, F]
    const float* w1 = (const float*)d_in[1];   // [F, H]
    const float* b1 = (const float*)d_in[2];   // [F, H]
    const float* w2 = (const float*)d_in[3];   // [F, H]
    const float* b2 = (const float*)d_in[4];   // [F]
    float* out      = (float*)d_out;           // [B,1] then [B,F] flat
    float* contribs = out + BATCH;

    dim3 grid(BATCH / ROWS);   // 256 blocks
    dim3 block(THREADS);       // 8 wave32s
    NAM_20126216750029_kernel<<<grid, block, SMEM_BYTES, stream>>>(
        x, w1, b1, w2, b2, out, contribs);
}